// GCN_69965017252459
// MI455X (gfx1250) — compile-verified
//
#include <hip/hip_runtime.h>
#include <hip/hip_bf16.h>

typedef float v2f __attribute__((ext_vector_type(2)));
typedef float v8f __attribute__((ext_vector_type(8)));

// ---------------------------------------------------------------------------
// Degree / normalization kernels
// ---------------------------------------------------------------------------
__global__ __launch_bounds__(256) void fill_deg(float* deg, int n) {
    int i = blockIdx.x * blockDim.x + threadIdx.x;
    if (i < n) deg[i] = 1.0f;  // self loop
}

__global__ __launch_bounds__(256) void count_deg(const int* dst, long long nE, float* deg) {
    long long e = (long long)blockIdx.x * blockDim.x + threadIdx.x;
    if (e < nE) unsafeAtomicAdd(&deg[dst[e]], 1.0f);
}

__global__ __launch_bounds__(256) void make_dinv(float* deg, int n) {
    int i = blockIdx.x * blockDim.x + threadIdx.x;
    if (i < n) deg[i] = __frsqrt_rn(deg[i]);  // deg >= 1 always
}

// ---------------------------------------------------------------------------
// WMMA fp32 GEMM:  C[M,N] = op(A)[M,K] @ W[K,N]   (op = identity or relu)
// One 16x16 output tile per wave; V_WMMA_F32_16X16X4_F32, K stepped by 4.
// A-frag layout: lane l holds A[m][k0], A[m][k0+1], m = l%16, k0 = (l/16)*2.
// B-frag layout: lane l holds B[k0][n], B[k0+1][n],  n = l%16, k0 = (l/16)*2.
// C/D layout:    VGPR r, lane l -> row r + 8*(l/16), col l%16.
// ---------------------------------------------------------------------------
template <int K, int N, bool RELU>
__global__ __launch_bounds__(256) void wmma_gemm(const float* __restrict__ A,
                                                 const float* __restrict__ W,
                                                 float* __restrict__ C,
                                                 int M) {
    const int lane   = threadIdx.x & 31;
    const int waveId = threadIdx.x >> 5;
    const int tilesN = N / 16;
    const int tile   = blockIdx.x * 8 + waveId;
    const int totalTiles = (M / 16) * tilesN;
    if (tile >= totalTiles) return;  // wave-uniform: EXEC stays all-1s

    const int tm = tile / tilesN;
    const int tn = tile % tilesN;

    const int m    = tm * 16 + (lane & 15);
    const int n    = tn * 16 + (lane & 15);
    const int koff = (lane >> 4) * 2;

    v8f acc = {};
    const float* __restrict__ arow = A + (size_t)m * K;
    #pragma unroll 4
    for (int k = 0; k < K; k += 4) {
        v2f a = *(const v2f*)(arow + k + koff);  // 8B aligned (k+koff even)
        if (RELU) {
            a.x = fmaxf(a.x, 0.0f);
            a.y = fmaxf(a.y, 0.0f);
        }
        v2f b;
        b.x = W[(size_t)(k + koff) * N + n];
        b.y = W[(size_t)(k + koff + 1) * N + n];
        acc = __builtin_amdgcn_wmma_f32_16x16x4_f32(
            /*neg_a=*/false, a, /*neg_b=*/false, b,
            /*c_mod=*/(short)0, acc, /*reuse_a=*/false, /*reuse_b=*/false);
    }

    const int rbase = tm * 16 + 8 * (lane >> 4);
    const int ccol  = tn * 16 + (lane & 15);
    #pragma unroll
    for (int r = 0; r < 8; ++r)
        C[(size_t)(rbase + r) * N + ccol] = acc[r];
}

// ---------------------------------------------------------------------------
// Aggregation init: out[i,f] = h[i,f]*dinv[i]^2 + bias[f]   (self loop + bias)
// ---------------------------------------------------------------------------
template <int F>
__global__ __launch_bounds__(256) void agg_init(const float* __restrict__ h,
                                                const float* __restrict__ dinv,
                                                const float* __restrict__ bias,
                                                float* __restrict__ out, int n) {
    long long t = (long long)blockIdx.x * blockDim.x + threadIdx.x;
    if (t >= (long long)n * F) return;
    int i = (int)(t / F);
    int f = (int)(t % F);
    float di = dinv[i];
    out[t] = h[t] * di * di + bias[f];
}

// ---------------------------------------------------------------------------
// Edge scatter: out[dst] += h[src] * dinv[src]*dinv[dst]
// F/4 threads per edge, float4 gather, native f32 atomic scatter.
// ---------------------------------------------------------------------------
template <int F>
__global__ __launch_bounds__(256) void edge_scatter(const float* __restrict__ h,
                                                    const int* __restrict__ src,
                                                    const int* __restrict__ dst,
                                                    const float* __restrict__ dinv,
                                                    float* __restrict__ out,
                                                    long long nE) {
    const int CH = F / 4;
    long long t = (long long)blockIdx.x * blockDim.x + threadIdx.x;
    long long e = t / CH;
    if (e >= nE) return;
    int c = (int)(t % CH) * 4;
    int s = src[e];
    int d = dst[e];
    float norm = dinv[s] * dinv[d];
    const float4 v = *(const float4*)(h + (size_t)s * F + c);
    float* o = out + (size_t)d * F + c;
    unsafeAtomicAdd(o + 0, v.x * norm);
    unsafeAtomicAdd(o + 1, v.y * norm);
    unsafeAtomicAdd(o + 2, v.z * norm);
    unsafeAtomicAdd(o + 3, v.w * norm);
}

// ---------------------------------------------------------------------------
// Mean pooling
// ---------------------------------------------------------------------------
__global__ __launch_bounds__(256) void zero_buf(float* p, int n) {
    int i = blockIdx.x * blockDim.x + threadIdx.x;
    if (i < n) p[i] = 0.0f;
}

template <int F>
__global__ __launch_bounds__(256) void pool_accum(const float* __restrict__ h,
                                                  const int* __restrict__ batch,
                                                  float* __restrict__ sums,
                                                  float* __restrict__ counts, int n) {
    long long t = (long long)blockIdx.x * blockDim.x + threadIdx.x;
    if (t >= (long long)n * F) return;
    int i = (int)(t / F);
    int f = (int)(t % F);
    int g = batch[i];
    unsafeAtomicAdd(&sums[(size_t)g * F + f], h[t]);
    if (f == 0) unsafeAtomicAdd(&counts[g], 1.0f);
}

template <int F>
__global__ __launch_bounds__(256) void pool_final(const float* __restrict__ sums,
                                                  const float* __restrict__ counts,
                                                  float* __restrict__ out, int nG) {
    int t = blockIdx.x * blockDim.x + threadIdx.x;
    if (t >= nG * F) return;
    int g = t / F;
    out[t] = sums[t] / fmaxf(counts[g], 1.0f);
}

// ---------------------------------------------------------------------------
// Host-side orchestration
// ---------------------------------------------------------------------------
static inline int ceil_div_ll(long long a, long long b) { return (int)((a + b - 1) / b); }

extern "C" void kernel_launch(void* const* d_in, const int* in_sizes, int n_in,
                              void* d_out, int out_size, void* d_ws, size_t ws_size,
                              hipStream_t stream) {
    constexpr int F1 = 128;  // input features
    constexpr int F2 = 256;  // hidden features
    constexpr int F3 = 128;  // output features

    const float* x    = (const float*)d_in[0];
    const int*   eidx = (const int*)d_in[1];
    const int*   batch= (const int*)d_in[2];
    const float* W1   = (const float*)d_in[3];
    const float* b1   = (const float*)d_in[4];
    const float* W2   = (const float*)d_in[5];
    const float* b2   = (const float*)d_in[6];
    float*       out  = (float*)d_out;

    const int       nNodes = in_sizes[0] / F1;          // 100000
    const long long nE     = (long long)in_sizes[1] / 2; // 3200000
    const int       nG     = out_size / F3;             // 64

    const int* src = eidx;
    const int* dst = eidx + nE;

    // Workspace layout (floats)
    float* ws    = (float*)d_ws;
    float* dinv  = ws;                               // nNodes
    float* H1    = dinv + nNodes;                    // nNodes*F2
    float* A1    = H1 + (size_t)nNodes * F2;         // nNodes*F2
    float* H2    = A1 + (size_t)nNodes * F2;         // nNodes*F3
    float* A2    = H1;                               // alias: H1 dead after layer-2 GEMM
    float* sums  = H2 + (size_t)nNodes * F3;         // nG*F3
    float* cnts  = sums + (size_t)nG * F3;           // nG

    const int B = 256;

    // 1) degree / dinv (shared by both layers)
    fill_deg<<<ceil_div_ll(nNodes, B), B, 0, stream>>>(dinv, nNodes);
    count_deg<<<ceil_div_ll(nE, B), B, 0, stream>>>(dst, nE, dinv);
    make_dinv<<<ceil_div_ll(nNodes, B), B, 0, stream>>>(dinv, nNodes);

    // 2) layer 1: H1 = x @ W1 (WMMA fp32)
    {
        long long tiles = (long long)(nNodes / 16) * (F2 / 16);
        wmma_gemm<F1, F2, false><<<ceil_div_ll(tiles, 8), B, 0, stream>>>(x, W1, H1, nNodes);
    }
    agg_init<F2><<<ceil_div_ll((long long)nNodes * F2, B), B, 0, stream>>>(H1, dinv, b1, A1, nNodes);
    edge_scatter<F2><<<ceil_div_ll(nE * (F2 / 4), B), B, 0, stream>>>(H1, src, dst, dinv, A1, nE);

    // 3) layer 2: H2 = relu(A1) @ W2 (WMMA fp32, relu fused into A-load)
    {
        long long tiles = (long long)(nNodes / 16) * (F3 / 16);
        wmma_gemm<F2, F3, true><<<ceil_div_ll(tiles, 8), B, 0, stream>>>(A1, W2, H2, nNodes);
    }
    agg_init<F3><<<ceil_div_ll((long long)nNodes * F3, B), B, 0, stream>>>(H2, dinv, b2, A2, nNodes);
    edge_scatter<F3><<<ceil_div_ll(nE * (F3 / 4), B), B, 0, stream>>>(H2, src, dst, dinv, A2, nE);

    // 4) global mean pool
    zero_buf<<<ceil_div_ll(nG * F3 + nG, B), B, 0, stream>>>(sums, nG * F3 + nG);
    pool_accum<F3><<<ceil_div_ll((long long)nNodes * F3, B), B, 0, stream>>>(A2, batch, sums, cnts, nNodes);
    pool_final<F3><<<ceil_div_ll(nG * F3, B), B, 0, stream>>>(sums, cnts, out, nG);
}